// ComputeHistograms_71159018160701
// MI455X (gfx1250) — compile-verified
//
#include <hip/hip_runtime.h>
#include <stdint.h>

// ---- problem constants (from reference setup) ----
#define WSZ    8
#define BINS   256
#define NB     8              // batch
#define NC     3              // channels
#define HH     1024
#define WW_    1024
#define NH     (HH / WSZ)     // 128
#define NW     (WW_ / WSZ)    // 128
#define PLANES (NB * NC)      // 24 image planes per window
#define WIN_ELEMS (PLANES * WSZ * WSZ)  // 1536 floats per window

typedef unsigned int v4u __attribute__((ext_vector_type(4)));
typedef int          v8i __attribute__((ext_vector_type(8)));
typedef int          v4i __attribute__((ext_vector_type(4)));

__global__ __launch_bounds__(256)
void ComputeHistograms_kernel(const float* __restrict__ x, float* __restrict__ out) {
    __shared__ float    win[WIN_ELEMS];   // 6 KB window tile (TDM destination)
    __shared__ unsigned bins[BINS];       // 1 KB privatized histogram

    const int tid = threadIdx.x;
    const int wid = blockIdx.x;
    const int wh  = wid / NW;
    const int ww  = wid % NW;

    bins[tid] = 0u;

#if __has_builtin(__builtin_amdgcn_tensor_load_to_lds) && __has_builtin(__builtin_amdgcn_s_wait_tensorcnt)
    if (tid < 32) {   // wave 0 issues the tensor DMA (uniform SGPR descriptor)
        // Tile start inside the tensor: x + (wh*8)*W + ww*8 (elements)
        const uint64_t gaddr = (uint64_t)(uintptr_t)x
                             + ((uint64_t)(wh * WSZ) * (uint64_t)WW_ + (uint64_t)(ww * WSZ)) * 4ull;
        // LDS byte offset of destination (generic LDS addr keeps offset in low 32 bits)
        const unsigned lds_off = (unsigned)(uintptr_t)(&win[0]);

        // ---- D# group 0 (128b): count=1 | lds_addr | global_addr[56:0] | type=2 ----
        v4u g0;
        g0.x = 1u;                                          // count=1, is_restore=0, gather=0
        g0.y = lds_off;                                     // lds_addr
        g0.z = (unsigned)(gaddr & 0xFFFFFFFFull);           // global_addr[31:0]
        g0.w = (unsigned)((gaddr >> 32) & 0x01FFFFFFull)    // global_addr[56:32]
             | (2u << 30);                                  // type = 2 ("image")

        // ---- D# group 1 (256b) ----
        // dim0 = x (width W), dim1 = y (rows, stride W), dim2 = plane (stride H*W)
        const uint64_t dim1_stride = (uint64_t)HH * (uint64_t)WW_;  // 1,048,576 elems
        v8i g1;
        g1[0] = (int)(2u << 16);                                   // wg_mask=0, data_size=2 (4B)
        g1[1] = (int)(((unsigned)WW_ & 0xFFFFu) << 16);            // tensor_dim0[15:0]
        g1[2] = (int)(((unsigned)WW_ >> 16)
              | (((unsigned)HH & 0xFFFFu) << 16));                 // dim0[31:16] | dim1[15:0]
        g1[3] = (int)(((unsigned)HH >> 16)
              | ((unsigned)WSZ << 16));                            // dim1[31:16] | tile_dim0=8
        g1[4] = (int)((unsigned)WSZ | ((unsigned)PLANES << 16));   // tile_dim1=8 | tile_dim2=24
        g1[5] = (int)WW_;                                          // tensor_dim0_stride[31:0]
        g1[6] = (int)(((unsigned)0)                                // dim0_stride[47:32]=0
              | (((unsigned)(dim1_stride & 0xFFFFull)) << 16));    // dim1_stride[15:0]
        g1[7] = (int)(dim1_stride >> 16);                          // dim1_stride[47:16]

        // ---- D# group 2 (128b, normal mode) ----
        v4i g2;
        g2[0] = PLANES;   // tensor_dim2 = 24
        g2[1] = 1;        // tensor_dim3 (unused)
        g2[2] = 0;        // tensor_dim2_stride[31:0] (unused, tile_dim3=0)
        g2[3] = 0;        // dim2_stride[47:32] | tile_dim3=0

        // ---- D# group 3 (128b): unused dims ----
        v4i g3 = {0, 0, 0, 0};

        // ---- arg 5: clang-23 6-arg form; extra int32x8 (VADDR4 slot is unused/NULL per ISA) ----
        v8i g4 = {0, 0, 0, 0, 0, 0, 0, 0};

        __builtin_amdgcn_tensor_load_to_lds(g0, g1, g2, g3, g4, 0);
        __builtin_amdgcn_s_wait_tensorcnt(0);
    }
#else
    // Fallback: cooperative global->LDS staging (coalesced float loads)
    for (int e = tid; e < WIN_ELEMS; e += 256) {
        const int plane = e >> 6;              // /64
        const int r     = (e >> 3) & 7;        // row in window
        const int c     = e & 7;               // col in window
        win[e] = x[(size_t)plane * HH * WW_ + (size_t)(wh * WSZ + r) * WW_ + (ww * WSZ + c)];
    }
#endif
    __syncthreads();

    // 6 elements per thread, stride-256 -> conflict-free LDS reads
    #pragma unroll
    for (int k = 0; k < WIN_ELEMS / 256; ++k) {
        const float v = win[tid + 256 * k];
        if (v >= 0.0f && v <= 1.0f) {                // torch.histc drops out-of-range
            int idx = (int)floorf(v * 256.0f);       // same f32 math as reference
            idx = idx > (BINS - 1) ? (BINS - 1) : idx;  // v==1.0 -> last bin
            atomicAdd(&bins[idx], 1u);               // ds_add_u32
        }
    }
    __syncthreads();

    // Broadcast histogram to all 8 batch slots: out[b][wh][ww][tid]
    const float h = (float)bins[tid];
    #pragma unroll
    for (int b = 0; b < NB; ++b) {
        out[(((size_t)b * NH + wh) * NW + ww) * BINS + tid] = h;
    }
}

extern "C" void kernel_launch(void* const* d_in, const int* in_sizes, int n_in,
                              void* d_out, int out_size, void* d_ws, size_t ws_size,
                              hipStream_t stream) {
    const float* x   = (const float*)d_in[0];
    float*       out = (float*)d_out;
    (void)in_sizes; (void)n_in; (void)d_ws; (void)ws_size; (void)out_size;

    ComputeHistograms_kernel<<<NH * NW, 256, 0, stream>>>(x, out);
}